// RecurrentGCN_26164940767929
// MI455X (gfx1250) — compile-verified
//
#include <hip/hip_runtime.h>
#include <math.h>

typedef __attribute__((ext_vector_type(2))) float v2f;
typedef __attribute__((ext_vector_type(8))) float v8f;

#define N_NODES 20
#define F_IN_C  8
#define KDIM    (N_NODES * F_IN_C)   // 160
#define NCOL    80                   // col = n*4 + o*2 + s ; s=0 -> R, s=1 -> HC
#define NTILES  (NCOL / 16)          // 5

// ------------------------------------------------------------------
// Prep kernel: fold all batch-independent structure into one combined
// weight matrix G (KDIM x NCOL, pair-interleaved over K) + bias (NCOL).
//   col = n*4 + o*2 + s :
//     s=0: R preact    -> gate_w output index o+2, gate_b
//     s=1: HC preact   -> upd_w output index o,   upd_b
//   G[(m*8+c), col] = delta(n,m)*W[n,0,c,*] + A[n,m]*W[n,1,c,*]
//   W[n,k,c,*] = sum_d e[n,d]*pool_w[d,k,c,*],  A = softmax(relu(e e^T))
// Stored interleaved: Gp[(kpair*NCOL + col)*2 + (krow&1)], kpair = krow/2,
// so the main kernel's B fragment is one aligned 8-byte LDS load.
// ------------------------------------------------------------------
__global__ void rgcn_prep_kernel(const float* __restrict__ e,
                                 const float* __restrict__ gate_w,
                                 const float* __restrict__ gate_b,
                                 const float* __restrict__ upd_w,
                                 const float* __restrict__ upd_b,
                                 float* __restrict__ Gp,
                                 float* __restrict__ bias)
{
    __shared__ float sA[N_NODES * N_NODES];
    const int tid = threadIdx.x;

    if (tid < N_NODES) {
        float v[N_NODES];
        float mx = 0.0f;  // relu keeps values >= 0
        for (int m = 0; m < N_NODES; ++m) {
            float d = 0.f;
            for (int dd = 0; dd < 4; ++dd) d += e[tid * 4 + dd] * e[m * 4 + dd];
            d = fmaxf(d, 0.f);
            v[m] = d;
            mx = fmaxf(mx, d);
        }
        float s = 0.f;
        for (int m = 0; m < N_NODES; ++m) { v[m] = __expf(v[m] - mx); s += v[m]; }
        const float inv = 1.f / s;
        for (int m = 0; m < N_NODES; ++m) sA[tid * N_NODES + m] = v[m] * inv;
    }
    __syncthreads();

    for (int idx = tid; idx < KDIM * NCOL; idx += (int)blockDim.x) {
        const int col  = idx % NCOL;
        const int krow = idx / NCOL;
        const int m    = krow / F_IN_C;
        const int c    = krow % F_IN_C;
        const int n    = col >> 2;
        const int o    = (col >> 1) & 1;
        const int s    = col & 1;
        float w0 = 0.f, w1 = 0.f;
        if (s == 0) {
            const int oo = o + 2;                       // R = gate outputs 2..3
            for (int d = 0; d < 4; ++d) {
                w0 += e[n * 4 + d] * gate_w[((d * 2 + 0) * 10 + c) * 4 + oo];
                w1 += e[n * 4 + d] * gate_w[((d * 2 + 1) * 10 + c) * 4 + oo];
            }
        } else {
            for (int d = 0; d < 4; ++d) {
                w0 += e[n * 4 + d] * upd_w[((d * 2 + 0) * 10 + c) * 2 + o];
                w1 += e[n * 4 + d] * upd_w[((d * 2 + 1) * 10 + c) * 2 + o];
            }
        }
        const float val = (m == n ? w0 : 0.f) + sA[n * N_NODES + m] * w1;
        // pair-interleaved store for single-b64 LDS fragment loads
        Gp[(((krow >> 1) * NCOL) + col) * 2 + (krow & 1)] = val;
    }

    for (int col = tid; col < NCOL; col += (int)blockDim.x) {
        const int n = col >> 2;
        const int o = (col >> 1) & 1;
        const int s = col & 1;
        float bv = 0.f;
        if (s == 0) {
            for (int d = 0; d < 4; ++d) bv += e[n * 4 + d] * gate_b[d * 4 + o + 2];
        } else {
            for (int d = 0; d < 4; ++d) bv += e[n * 4 + d] * upd_b[d * 2 + o];
        }
        bias[col] = bv;
    }
}

// ------------------------------------------------------------------
// Main kernel: HBM-bound streaming GEMM (B x 160) * (160 x 80) on the
// fp32 matrix units (v_wmma_f32_16x16x4_f32) + fused GRU epilogue.
// 8 waves/block; each wave owns a 16-row batch tile and 5 accumulators.
// ------------------------------------------------------------------
__global__ __launch_bounds__(256) void rgcn_main_kernel(
    const float* __restrict__ x,
    const float* __restrict__ Gp,
    const float* __restrict__ bias,
    const float* __restrict__ lin_w,
    const float* __restrict__ lin_b,
    float* __restrict__ y,
    int Btot)
{
    __shared__ float sG[KDIM * NCOL];   // pair-interleaved [kpair][col][2]
    __shared__ float sBias[NCOL];

    const int tid = threadIdx.x;
    for (int i = tid; i < KDIM * NCOL; i += 256) sG[i] = Gp[i];
    if (tid < NCOL) sBias[tid] = bias[tid];
    __syncthreads();

    const int wave = tid >> 5;
    const int lane = tid & 31;
    const int b0   = blockIdx.x * 128 + wave * 16;
    if (b0 + 16 > Btot) return;   // wave-uniform: EXEC stays all-1 for WMMA

    const int lcol  = lane & 15;  // M row (A) / N col (B,D) within 16x16 tile
    const int khalf = lane >> 4;  // which K pair this lane holds

    const float* xrow = x + (size_t)(b0 + lcol) * KDIM;

    v8f acc[NTILES] = {};

    for (int k0 = 0; k0 < KDIM; k0 += 4) {
        const int kk    = k0 + 2 * khalf;     // K index of VGPR0 for this lane
        const int pbase = (k0 >> 1) + khalf;  // k-pair index in interleaved G
        v2f a;
        a.x = xrow[kk];
        a.y = xrow[kk + 1];
        #pragma unroll
        for (int t = 0; t < NTILES; ++t) {
            const v2f bf = *reinterpret_cast<const v2f*>(
                &sG[(pbase * NCOL + t * 16 + lcol) * 2]);
            acc[t] = __builtin_amdgcn_wmma_f32_16x16x4_f32(
                false, a, false, bf, (short)0, acc[t], false, false);
        }
    }

    // Epilogue. col = n*4 + o*2 + s : even lanes hold R preact, odd lanes the
    // matching HC preact (adjacent lane). H = (1-sigmoid(R))*tanh(HC);
    // y[b,n] = relu(H[n,0])*lw0 + relu(H[n,1])*lw1 + lb, reduced via xor-2.
    const float lw0 = lin_w[0], lw1 = lin_w[1], lb = lin_b[0];

    #pragma unroll
    for (int t = 0; t < NTILES; ++t) {
        const int   col = t * 16 + lcol;           // 0..79, no padding
        const float lw  = ((col >> 1) & 1) ? lw1 : lw0;
        const float bv  = sBias[col];
        #pragma unroll
        for (int r = 0; r < 8; ++r) {
            const float pre  = acc[t][r] + bv;
            const float part = __shfl_xor(pre, 1, 32);     // partner preact
            const float rp   = (col & 1) ? part : pre;     // R preact
            const float hp   = (col & 1) ? pre : part;     // HC preact
            const float R    = 1.f / (1.f + __expf(-rp));
            const float HC   = 1.f - 2.f / (__expf(2.f * hp) + 1.f);  // tanh
            const float H    = (1.f - R) * HC;
            const float p    = fmaxf(H, 0.f) * lw;         // same on lane pair
            const float q    = p + __shfl_xor(p, 2, 32);   // (n,0)+(n,1)
            if ((col & 3) == 0) {
                y[(size_t)(b0 + r + 8 * khalf) * N_NODES + (col >> 2)] = q + lb;
            }
        }
    }
}

extern "C" void kernel_launch(void* const* d_in, const int* in_sizes, int n_in,
                              void* d_out, int out_size, void* d_ws, size_t ws_size,
                              hipStream_t stream) {
    const float* x      = (const float*)d_in[0];
    const float* e      = (const float*)d_in[1];
    const float* gate_w = (const float*)d_in[2];
    const float* gate_b = (const float*)d_in[3];
    const float* upd_w  = (const float*)d_in[4];
    const float* upd_b  = (const float*)d_in[5];
    const float* lin_w  = (const float*)d_in[6];
    const float* lin_b  = (const float*)d_in[7];
    float* y = (float*)d_out;

    const int Btot = in_sizes[0] / KDIM;   // 131072

    float* Gp   = (float*)d_ws;            // KDIM*NCOL floats (interleaved)
    float* bias = Gp + KDIM * NCOL;        // NCOL floats

    rgcn_prep_kernel<<<1, 256, 0, stream>>>(e, gate_w, gate_b, upd_w, upd_b, Gp, bias);

    const int blocks = (Btot + 127) / 128; // 16 rows/wave * 8 waves
    rgcn_main_kernel<<<blocks, 256, 0, stream>>>(x, Gp, bias, lin_w, lin_b, y, Btot);
}